// IGConv_21019569947058
// MI455X (gfx1250) — compile-verified
//
#include <hip/hip_runtime.h>
#include <hip/hip_bf16.h>

// ---------------------------------------------------------------------------
// Implicit GNN fixed point:  z <- tanh( A @ (z @ wp) + bias ), 301 iterations.
// wp = column-wise L1 projection of w (norm_inf, kappa = 0.9).
// D = 64 fixed. CSR built once per call (deterministic per-row edge order).
// ---------------------------------------------------------------------------

typedef float v2f __attribute__((ext_vector_type(2)));
typedef float v8f __attribute__((ext_vector_type(8)));

#define KAPPA 0.9f
#define DFEAT 64
#define PROPS 300

// ---------------- wp = norm_inf(w) : one thread per column ------------------
__global__ void proj_kernel(const float* __restrict__ w, float* __restrict__ wp) {
    int col = blockIdx.x * blockDim.x + threadIdx.x;
    if (col >= DFEAT) return;
    float vals[DFEAT], a[DFEAT];
    float s = 0.f;
    for (int i = 0; i < DFEAT; ++i) {
        float v = w[i * DFEAT + col];
        vals[i] = v;
        float av = fabsf(v);
        a[i] = av;
        s += av;
    }
    // insertion sort ascending
    for (int i = 1; i < DFEAT; ++i) {
        float key = a[i];
        int j = i - 1;
        while (j >= 0 && a[j] > key) { a[j + 1] = a[j]; --j; }
        a[j + 1] = key;
    }
    float cl1 = s - KAPPA;
    float m = 0.f;
    if (cl1 > 0.f) {
        float excl = 0.f;
        for (int i = 0; i < DFEAT; ++i) {
            float minus = (cl1 - excl) / (float)(DFEAT - i);
            if (minus <= a[i]) { m = minus; break; }
            excl += a[i];
        }
    }
    for (int i = 0; i < DFEAT; ++i) {
        float v = vals[i];
        float r = fabsf(v) - m;
        r = r > 0.f ? r : 0.f;
        wp[i * DFEAT + col] = (v > 0.f) ? r : (v < 0.f ? -r : 0.f);
    }
}

// ---------------- CSR build (once per call) --------------------------------
__global__ void zero_kernel(int* __restrict__ p, int n) {
    int i = blockIdx.x * blockDim.x + threadIdx.x;
    if (i < n) p[i] = 0;
}

__global__ void count_kernel(const int* __restrict__ src, int* __restrict__ counts, int E) {
    int e = blockIdx.x * blockDim.x + threadIdx.x;
    if (e < E) atomicAdd(&counts[src[e]], 1);
}

// single block, 256 threads: exclusive scan of counts -> row_start, cursor
__global__ void scan_kernel(const int* __restrict__ counts, int* __restrict__ row_start,
                            int* __restrict__ cursor, int N) {
    __shared__ int sums[256];
    __shared__ int prefix[257];
    int t = threadIdx.x;
    int chunk = (N + 255) / 256;
    int beg = t * chunk;
    int end = beg + chunk; if (end > N) end = N;
    int s = 0;
    for (int i = beg; i < end; ++i) s += counts[i];
    sums[t] = s;
    __syncthreads();
    if (t == 0) {
        int acc = 0;
        for (int i = 0; i < 256; ++i) { prefix[i] = acc; acc += sums[i]; }
        prefix[256] = acc;
    }
    __syncthreads();
    int acc = prefix[t];
    for (int i = beg; i < end; ++i) {
        row_start[i] = acc;
        cursor[i] = acc;
        acc += counts[i];
    }
    if (t == 0) row_start[N] = prefix[256];
}

__global__ void fill_kernel(const int* __restrict__ src, int* __restrict__ cursor,
                            int* __restrict__ edge_idx, int E) {
    int e = blockIdx.x * blockDim.x + threadIdx.x;
    if (e < E) {
        int p = atomicAdd(&cursor[src[e]], 1);
        edge_idx[p] = e;
    }
}

// sort each row's edge ids ascending (deterministic order), then gather dst/w
__global__ void sortgather_kernel(const int* __restrict__ row_start, int* __restrict__ edge_idx,
                                  const int* __restrict__ edge_dst, const float* __restrict__ edge_w,
                                  int* __restrict__ sdst, float* __restrict__ sw, int N) {
    int r = blockIdx.x * blockDim.x + threadIdx.x;
    if (r >= N) return;
    int b = row_start[r], en = row_start[r + 1];
    for (int i = b + 1; i < en; ++i) {
        int key = edge_idx[i];
        int j = i - 1;
        while (j >= b && edge_idx[j] > key) { edge_idx[j + 1] = edge_idx[j]; --j; }
        edge_idx[j + 1] = key;
    }
    for (int k = b; k < en; ++k) {
        int e = edge_idx[k];
        sdst[k] = edge_dst[e];
        sw[k] = edge_w[e];
    }
}

// ---------------- y = z @ wp via V_WMMA_F32_16X16X4_F32 --------------------
// One wave computes a 16x64 output tile: 4 N-tiles x 16 K-steps (K=4 each).
// A layout (ISA 7.12.2, 32-bit A 16x4): lane&15 = row; half-wave selects K{0,1}
// vs K{2,3}; VGPR0 = K even, VGPR1 = K odd.  B (4x16) mirrors: lane&15 = col.
//
// LDS staging is PAIR-INTERLEAVED: element (k, c) at lwp[(k>>1)*128 + c*2 +
// (k&1)], so each B fragment {wp[kk][col], wp[kk+1][col]} is one aligned
// ds_load_b64 landing directly in an even VGPR pair (no shuffle moves).
__global__ void gemm_kernel(const float* __restrict__ z, const float* __restrict__ wp,
                            float* __restrict__ y, int nRowTiles) {
    __shared__ float lwp[DFEAT * DFEAT];
    for (int i = threadIdx.x; i < DFEAT * DFEAT; i += blockDim.x) {
        int k = i >> 6;   // row of wp (K index)
        int c = i & 63;   // col of wp (N index)
        lwp[(k >> 1) * 128 + c * 2 + (k & 1)] = wp[i];
    }
    __syncthreads();

    int wave = (blockIdx.x * blockDim.x + threadIdx.x) >> 5;
    if (wave >= nRowTiles) return;          // wave-uniform: EXEC stays all-1s
    int lane = threadIdx.x & 31;
    int half = lane >> 4;                   // 0 or 1
    int l15  = lane & 15;
    int rowBase = wave * 16;

    const float* zr = z + (size_t)(rowBase + l15) * DFEAT + half * 2;
    v2f a[16];
#pragma unroll
    for (int kt = 0; kt < 16; ++kt) {
        a[kt].x = zr[kt * 4];
        a[kt].y = zr[kt * 4 + 1];
    }

#pragma unroll
    for (int nt = 0; nt < 4; ++nt) {
        int col = nt * 16 + l15;
        v8f c = {};
#pragma unroll
        for (int kt = 0; kt < 16; ++kt) {
            int pair = kt * 2 + half;       // (kk>>1), kk = kt*4 + half*2
            v2f b = *(const v2f*)&lwp[pair * 128 + col * 2];
            c = __builtin_amdgcn_wmma_f32_16x16x4_f32(
                    /*neg_a=*/false, a[kt], /*neg_b=*/false, b,
                    /*c_mod=*/(short)0, c, /*reuse_a=*/false, /*reuse_b=*/false);
        }
        // C/D layout: VGPR v, lanes 0-15 -> M=v, lanes 16-31 -> M=v+8; N=l15
        float* yr = y + (size_t)rowBase * DFEAT + col;
#pragma unroll
        for (int v = 0; v < 8; ++v) {
            yr[(size_t)(v + half * 8) * DFEAT] = c[v];
        }
    }
}

// ------- z_out = tanh( segment_sum(ew * y[dst]) + bias ) : 1 wave / row -----
__global__ void spmm_kernel(const float* __restrict__ y, const int* __restrict__ row_start,
                            const int* __restrict__ sdst, const float* __restrict__ sw,
                            const float* __restrict__ bias, float* __restrict__ zout, int N) {
    int row  = (blockIdx.x * blockDim.x + threadIdx.x) >> 5;
    int lane = threadIdx.x & 31;
    if (row >= N) return;
    int beg = row_start[row], end = row_start[row + 1];
    float acc0 = 0.f, acc1 = 0.f;
    for (int k = beg; k < end; ++k) {
        int d = sdst[k];
        float wgt = sw[k];
        if (k + 1 < end)
            __builtin_prefetch(&y[(size_t)sdst[k + 1] * DFEAT], 0, 3);
        const float* yr = y + (size_t)d * DFEAT;
        acc0 = fmaf(wgt, yr[lane], acc0);
        acc1 = fmaf(wgt, yr[lane + 32], acc1);
    }
    const float* br = bias + (size_t)row * DFEAT;
    float* zr = zout + (size_t)row * DFEAT;
    zr[lane]      = tanhf(acc0 + br[lane]);
    zr[lane + 32] = tanhf(acc1 + br[lane + 32]);
}

// ---------------------------------------------------------------------------
extern "C" void kernel_launch(void* const* d_in, const int* in_sizes, int n_in,
                              void* d_out, int out_size, void* d_ws, size_t ws_size,
                              hipStream_t stream) {
    const int* edge_src = (const int*)d_in[0];
    const int* edge_dst = (const int*)d_in[1];
    const float* edge_w = (const float*)d_in[2];
    const float* bias   = (const float*)d_in[3];
    const float* w      = (const float*)d_in[4];
    const float* z0     = (const float*)d_in[5];

    const int E = in_sizes[0];
    const int N = in_sizes[3] / DFEAT;

    // workspace carve-up (256B aligned)
    char* ws = (char*)d_ws;
    size_t off = 0;
    auto carve = [&](size_t bytes) { char* p = ws + off; off = (off + bytes + 255) & ~size_t(255); return p; };
    float* wp        = (float*)carve(DFEAT * DFEAT * sizeof(float));
    float* ybuf      = (float*)carve((size_t)N * DFEAT * sizeof(float));
    float* zbuf      = (float*)carve((size_t)N * DFEAT * sizeof(float));
    int*   counts    = (int*)carve((size_t)N * sizeof(int));
    int*   cursor    = (int*)carve((size_t)N * sizeof(int));
    int*   row_start = (int*)carve((size_t)(N + 1) * sizeof(int));
    int*   edge_idx  = (int*)carve((size_t)E * sizeof(int));
    int*   sdst      = (int*)carve((size_t)E * sizeof(int));
    float* sw        = (float*)carve((size_t)E * sizeof(float));

    // ---- once-per-call preprocessing ----
    proj_kernel<<<1, 64, 0, stream>>>(w, wp);
    zero_kernel<<<(N + 255) / 256, 256, 0, stream>>>(counts, N);
    count_kernel<<<(E + 255) / 256, 256, 0, stream>>>(edge_src, counts, E);
    scan_kernel<<<1, 256, 0, stream>>>(counts, row_start, cursor, N);
    fill_kernel<<<(E + 255) / 256, 256, 0, stream>>>(edge_src, cursor, edge_idx, E);
    sortgather_kernel<<<(N + 255) / 256, 256, 0, stream>>>(row_start, edge_idx,
                                                           edge_dst, edge_w, sdst, sw, N);

    // z = z0
    hipMemcpyAsync(zbuf, z0, (size_t)N * DFEAT * sizeof(float),
                   hipMemcpyDeviceToDevice, stream);

    // ---- fixed-point loop: tol = 0.001*max|z0| = 0 -> full 300 iters,
    //      plus the final recompute = 301 total; last writes d_out ----
    const int nRowTiles = (N + 15) / 16;
    const int gemmBlocks = (nRowTiles + 7) / 8;          // 8 waves (256 thr) / block
    const int spmmBlocks = (int)(((size_t)N * 32 + 255) / 256); // 1 wave / row

    for (int it = 0; it <= PROPS; ++it) {
        gemm_kernel<<<gemmBlocks, 256, 0, stream>>>(zbuf, wp, ybuf, nRowTiles);
        float* zout = (it == PROPS) ? (float*)d_out : zbuf;
        spmm_kernel<<<spmmBlocks, 256, 0, stream>>>(ybuf, row_start, sdst, sw,
                                                    bias, zout, N);
    }
}